// SpaceSimilarityLossV2_9491877724863
// MI455X (gfx1250) — compile-verified
//
#include <hip/hip_runtime.h>

// ---------------------------------------------------------------------------
// Contrastive/space-similarity loss, fused:
//   S = X X^T per batch via V_WMMA_F32_16X16X4_F32 (fp32-exact),
//   dist = sqrt(max(|xi|^2 + |xj|^2 - 2 S, 0)),
//   loss = mean( same ? max(d-1,0) : max(10-d,0) )
// B=8, N=4096, D=64.  128x128 tiles, 8 waves/block, epilogue fused in regs.
// ---------------------------------------------------------------------------

typedef float v2f __attribute__((ext_vector_type(2)));
typedef float v8f __attribute__((ext_vector_type(8)));

#define BATCH 8
#define NPTS  4096
#define DIM   64
#define TILE  128
#define LDSTR 68   // 64 + 4 floats pad: keeps b64 fragment reads conflict-free,
                   // and 68*4=272 bytes is 16B-aligned for float4 stores.

__global__ void ssl_init_kernel(double* accum) {
    if (threadIdx.x == 0) accum[0] = 0.0;
}

__global__ __launch_bounds__(256)
void ssl_tile_kernel(const float* __restrict__ X, const int* __restrict__ T,
                     double* __restrict__ accum) {
    __shared__ float lds_a[TILE * LDSTR];
    __shared__ float lds_b[TILE * LDSTR];
    __shared__ float x2a[TILE];
    __shared__ float x2b[TILE];
    __shared__ int   tga[TILE];
    __shared__ int   tgb[TILE];
    __shared__ float wave_part[8];

    const int b       = blockIdx.z;
    const int rowBase = blockIdx.y * TILE;
    const int colBase = blockIdx.x * TILE;
    const int t       = threadIdx.x;

    const float* Xb = X + (size_t)b * NPTS * DIM;

    // ---- stage 128x64 A-rows and B-rows into LDS (float4, fully coalesced)
    const float4* srcA = (const float4*)(Xb + (size_t)rowBase * DIM);
    const float4* srcB = (const float4*)(Xb + (size_t)colBase * DIM);
#pragma unroll
    for (int it = 0; it < 8; ++it) {
        int idx = t + it * 256;        // 0..2047 float4s; D=64 -> 16 f4 per row
        int r   = idx >> 4;
        int c4  = idx & 15;
        float4 va = srcA[idx];
        float4 vb = srcB[idx];
        *(float4*)&lds_a[r * LDSTR + c4 * 4] = va;
        *(float4*)&lds_b[r * LDSTR + c4 * 4] = vb;
    }
    if (t < TILE) tga[t]        = T[(size_t)b * NPTS + rowBase + t];
    else          tgb[t - TILE] = T[(size_t)b * NPTS + colBase + (t - TILE)];
    __syncthreads();

    // ---- row norms from the staged slabs (no extra global traffic)
    {
        int r = t & (TILE - 1);
        const float* src = (t < TILE) ? &lds_a[r * LDSTR] : &lds_b[r * LDSTR];
        float s = 0.f;
#pragma unroll
        for (int k = 0; k < DIM; ++k) s += src[k] * src[k];
        if (t < TILE) x2a[r] = s; else x2b[r] = s;
    }
    __syncthreads();

    // ---- WMMA main loop: each wave owns a 32x64 region (2x4 16x16 subtiles)
    const int wave = t >> 5;
    const int lane = t & 31;
    const int mOff = (wave >> 1) * 32;   // 0,32,64,96
    const int nOff = (wave & 1) * 64;    // 0,64
    const int m    = lane & 15;
    const int kh   = lane >> 4;          // 0/1: K-half per ISA 16x4 f32 layout

    v8f c[2][4] = {};

#pragma unroll 4
    for (int kb = 0; kb < DIM; kb += 4) {
        const int ko = kb + 2 * kh;      // VGPR0 holds K=ko, VGPR1 holds K=ko+1
        v2f afrag[2];
        v2f bfrag[4];
#pragma unroll
        for (int i = 0; i < 2; ++i)
            afrag[i] = *(const v2f*)&lds_a[(mOff + i * 16 + m) * LDSTR + ko];
#pragma unroll
        for (int j = 0; j < 4; ++j)
            bfrag[j] = *(const v2f*)&lds_b[(nOff + j * 16 + m) * LDSTR + ko];
#pragma unroll
        for (int i = 0; i < 2; ++i)
#pragma unroll
            for (int j = 0; j < 4; ++j)
                c[i][j] = __builtin_amdgcn_wmma_f32_16x16x4_f32(
                    /*neg_a=*/false, afrag[i],
                    /*neg_b=*/false, bfrag[j],
                    /*c_mod=*/(short)0, c[i][j],
                    /*reuse_a=*/false, /*reuse_b=*/false);
    }

    // ---- fused epilogue on accumulator fragments
    // C/D layout: VGPR r -> M = r + 8*(lane>=16), N = lane&15
    float lsum = 0.f;
#pragma unroll
    for (int j = 0; j < 4; ++j) {
        const int jl  = nOff + j * 16 + m;
        const float x2j = x2b[jl];
        const int   tj  = tgb[jl];
#pragma unroll
        for (int i = 0; i < 2; ++i) {
#pragma unroll
            for (int r = 0; r < 8; ++r) {
                const int il = mOff + i * 16 + r + 8 * kh;
                float sq = x2a[il] + x2j - 2.0f * c[i][j][r];
                sq = fmaxf(sq, 0.0f);
                float d = __builtin_sqrtf(sq);
                float loss = (tga[il] == tj) ? fmaxf(d - 1.0f, 0.0f)
                                             : fmaxf(10.0f - d, 0.0f);
                lsum += loss;
            }
        }
    }

    // ---- wave32 reduction, then block reduction, then one f64 atomic
#pragma unroll
    for (int off = 16; off > 0; off >>= 1)
        lsum += __shfl_down(lsum, off, 32);
    if (lane == 0) wave_part[wave] = lsum;
    __syncthreads();
    if (t == 0) {
        float s = 0.f;
#pragma unroll
        for (int w = 0; w < 8; ++w) s += wave_part[w];
        atomicAdd(accum, (double)s);   // global_atomic_add_f64
    }
}

__global__ void ssl_finalize_kernel(const double* __restrict__ accum,
                                    float* __restrict__ out) {
    if (threadIdx.x == 0) {
        const double cnt = (double)BATCH * (double)NPTS * (double)NPTS;
        out[0] = (float)(accum[0] / cnt);
    }
}

extern "C" void kernel_launch(void* const* d_in, const int* in_sizes, int n_in,
                              void* d_out, int out_size, void* d_ws, size_t ws_size,
                              hipStream_t stream) {
    const float* X = (const float*)d_in[0];   // [8,4096,64] f32
    const int*   T = (const int*)d_in[1];     // [8,4096] i32
    float* out     = (float*)d_out;           // scalar f32
    double* accum  = (double*)d_ws;           // 8-byte f64 accumulator

    ssl_init_kernel<<<1, 32, 0, stream>>>(accum);

    dim3 grid(NPTS / TILE, NPTS / TILE, BATCH);   // (32, 32, 8)
    ssl_tile_kernel<<<grid, 256, 0, stream>>>(X, T, accum);

    ssl_finalize_kernel<<<1, 32, 0, stream>>>(accum, out);
}